// SIMP_ADMM_22419729285763
// MI455X (gfx1250) — compile-verified
//
#include <hip/hip_runtime.h>
#include <math.h>

// Problem constants (match reference)
#define NME    262144
#define KE     64
#define NDOF   524288
#define EMIN_F 1e-9f
#define EMAX_F 1.0f
#define VOLFRAC_F 0.4f

typedef __attribute__((ext_vector_type(2))) float v2f;
typedef __attribute__((ext_vector_type(8))) float v8f;

// ---------------------------------------------------------------------------
// xor-16 lane swap + add. ds_swizzle_b32 SWAPX16: group-of-32 mode,
// offset = (xor_mask 0x10 << 10) | (or_mask 0 << 5) | (and_mask 0x1f) = 0x401F.
// Single DS op instead of ds_bpermute + bounds cndmask.
// ---------------------------------------------------------------------------
__device__ __forceinline__ float xor16_add(float s) {
#if __has_builtin(__builtin_amdgcn_ds_swizzle)
  int o = __builtin_amdgcn_ds_swizzle(__float_as_int(s), 0x401F);
  return s + __int_as_float(o);
#else
  return s + __shfl_xor(s, 16, 32);
#endif
}

// ---------------------------------------------------------------------------
// Wave32 sum reduction using the CDNA5 matrix pipe:
// V_WMMA_F32_16X16X4_F32 with B = all-ones computes D[m][n] = rowsum_m(A).
// A holds the wave's 32 partials (1 per lane, 2nd slot zero). Lanes 0-15 end up with rowsums
// of rows 0-7 in their 8 D VGPRs, lanes 16-31 with rows 8-15; one xor-16
// swap completes the 32-lane sum. EXEC must be all-ones at the call site
// (all callers keep uniform control flow).
// ---------------------------------------------------------------------------
__device__ __forceinline__ float wave_sum32(float x) {
#if __has_builtin(__builtin_amdgcn_wmma_f32_16x16x4_f32)
  v2f a; a[0] = x;    a[1] = 0.0f;
  v2f b; b[0] = 1.0f; b[1] = 1.0f;
  v8f c = {};
  v8f d = __builtin_amdgcn_wmma_f32_16x16x4_f32(
      /*neg_a=*/false, a, /*neg_b=*/false, b,
      /*c_mod=*/(short)0, c, /*reuse_a=*/false, /*reuse_b=*/false);
  float s = ((d[0] + d[1]) + (d[2] + d[3])) + ((d[4] + d[5]) + (d[6] + d[7]));
  return xor16_add(s);     // combine the two 8-row halves
#else
  #pragma unroll
  for (int off = 16; off > 0; off >>= 1) x += __shfl_xor(x, off, 32);
  return x;
#endif
}

// sigmoid with v_exp_f32 + v_rcp_f32 (avoids the IEEE v_div_fmas expansion)
__device__ __forceinline__ float sigmoid_f(float x) {
#if __has_builtin(__builtin_amdgcn_rcpf)
  return __builtin_amdgcn_rcpf(1.0f + __expf(-x));
#else
  return 1.0f / (1.0f + __expf(-x));
#endif
}

// --- zero Ku accumulator (NDOF floats) + the two scalar accumulators -------
__global__ void k_zero_ws(float4* __restrict__ Ku4, float* __restrict__ acc) {
  int i = blockIdx.x * blockDim.x + threadIdx.x;
  if (i < NDOF / 4) Ku4[i] = make_float4(0.f, 0.f, 0.f, 0.f);
  if (i == 0) { acc[0] = 0.f; acc[1] = 0.f; }
}

// --- sum of sigmoid(W_x) for the volume-fraction term ----------------------
__global__ void k_rho_sum(const float* __restrict__ Wx, float* __restrict__ acc) {
  int tid    = blockIdx.x * blockDim.x + threadIdx.x;
  int stride = gridDim.x * blockDim.x;
  float s = 0.f;
  for (int i = tid; i < NME; i += stride)   // uniform trip count: EXEC stays full
    s += sigmoid_f(Wx[i]);
  s = wave_sum32(s);
  if ((threadIdx.x & 31) == 0) atomicAdd(&acc[0], s);
}

// --- fused scale * gather * scatter-add: Ku[rows] += sK * u[cols] ----------
// One thread per 4 consecutive COO entries -> 128-bit coalesced loads.
// 16 threads share one element; sigmoid recompute is cheaper than a 1MB
// scale array, and the shared W_x read broadcasts from L0/L2.
__global__ void k_scatter(const float* __restrict__ Wx,
                          const float4* __restrict__ K4,
                          const int4* __restrict__ rows4,
                          const int4* __restrict__ cols4,
                          const float* __restrict__ u,
                          float* __restrict__ Ku) {
  int t = blockIdx.x * blockDim.x + threadIdx.x;   // [0, NME*KE/4)
  int e = t >> 4;                                  // element index
  float rho   = sigmoid_f(Wx[e]);
  float scale = EMIN_F + rho * rho * rho * (EMAX_F - EMIN_F);

  float4 k = K4[t];
  int4   r = rows4[t];
  int4   c = cols4[t];

  // gathers hit the 2MB u[] resident in L2; atomics are no-return f32 adds
  atomicAdd(&Ku[r.x], k.x * scale * u[c.x]);
  atomicAdd(&Ku[r.y], k.y * scale * u[c.y]);
  atomicAdd(&Ku[r.z], k.z * scale * u[c.z]);
  atomicAdd(&Ku[r.w], k.w * scale * u[c.w]);
}

// --- sum of (Ku - f)^2 ------------------------------------------------------
__global__ void k_norm(const float* __restrict__ Ku,
                       const float* __restrict__ f,
                       float* __restrict__ acc) {
  int tid    = blockIdx.x * blockDim.x + threadIdx.x;
  int stride = gridDim.x * blockDim.x;
  float s = 0.f;
  for (int i = tid; i < NDOF; i += stride) {     // uniform trip count
    __builtin_prefetch(&Ku[i + stride], 0, 0);   // global_prefetch_b8
    __builtin_prefetch(&f[i + stride], 0, 0);
    float d = Ku[i] - f[i];
    s = fmaf(d, d, s);
  }
  s = wave_sum32(s);
  if ((threadIdx.x & 31) == 0) atomicAdd(&acc[1], s);
}

// --- combine scalars --------------------------------------------------------
__global__ void k_final(const float* __restrict__ acc, float* __restrict__ out) {
  if (blockIdx.x == 0 && threadIdx.x == 0) {
    float mean = acc[0] * (1.0f / (float)NME);
    float vf   = mean - VOLFRAC_F;
    vf = vf > 0.f ? vf : 0.f;                    // relu
    out[0] = vf + sqrtf(acc[1]);                 // vofrac + gamma2 * ||Ku - f||
  }
}

extern "C" void kernel_launch(void* const* d_in, const int* in_sizes, int n_in,
                              void* d_out, int out_size, void* d_ws, size_t ws_size,
                              hipStream_t stream) {
  const float* Wx   = (const float*)d_in[0];               // [NME]
  const float* Ksep = (const float*)d_in[1];               // [NME*KE]
  const int*   idx  = (const int*)d_in[2];                 // [2, NME*KE]
  const float* u    = (const float*)d_in[3];               // [NDOF]
  const float* f    = (const float*)d_in[4];               // [NDOF]
  float*       out  = (float*)d_out;

  float* Ku  = (float*)d_ws;         // NDOF floats
  float* acc = Ku + NDOF;            // acc[0] = sum(rho), acc[1] = sum((Ku-f)^2)

  const int* rows = idx;
  const int* cols = idx + (size_t)NME * KE;

  // 1) zero accumulators (harness does not re-poison between replays)
  k_zero_ws<<<(NDOF / 4 + 255) / 256, 256, 0, stream>>>((float4*)Ku, acc);

  // 2) volume-fraction partial sum
  k_rho_sum<<<256, 256, 0, stream>>>(Wx, acc);

  // 3) fused SIMP scale + sparse matvec assembly (16.7M entries, 4/thread)
  const int n4 = (NME * KE) / 4;                 // 4,194,304 threads
  k_scatter<<<n4 / 256, 256, 0, stream>>>(Wx, (const float4*)Ksep,
                                          (const int4*)rows, (const int4*)cols,
                                          u, Ku);

  // 4) residual squared-norm partial sum
  k_norm<<<512, 256, 0, stream>>>(Ku, f, acc);

  // 5) scalar combine
  k_final<<<1, 64, 0, stream>>>(acc, out);
}